// Cross_stitch_21139829030962
// MI455X (gfx1250) — compile-verified
//
#include <hip/hip_runtime.h>

typedef __attribute__((ext_vector_type(2))) float v2f;
typedef __attribute__((ext_vector_type(8))) float v8f;

static constexpr int    kN     = 32;
static constexpr int    kC     = 256;
static constexpr int    kHW    = 64 * 64;                       // 4096 per (n,c) slab
static constexpr size_t kElems = (size_t)kN * kC * kHW;         // 33,554,432 per tensor

// Swap the two 16-lane halves of the wave: ds_swizzle group-of-32,
// xor_mask=0x10 -> swizzle(SWAP,16). Must run with full EXEC (reads from
// disabled lanes return zero), so it is only used in straight-line code.
__device__ __forceinline__ float swap16(float v) {
    return __int_as_float(__builtin_amdgcn_ds_swizzle(__float_as_int(v), 0x401F));
}

// One block = one (n,c) slab of 4096 contiguous floats in each tensor.
// 8 waves/block, each wave owns 512 positions, 32 per iteration.
//
// WMMA mapping (V_WMMA_F32_16X16X4_F32, D = A(16x4) x B(4x16) + 0):
//   A row0 = [w00 w01 0 0]  -> D row0 = out_a
//   A row1 = [w10 w11 0 0]  -> D row1 = out_b
//   A rows 2..15 and K2/K3 columns are zero.
// A/B f32 2-VGPR layout: VGPR0 = K0(lanes0-15)|K2(lanes16-31), VGPR1 = K1|K3.
// Consequences used below:
//   * garbage in B's K2/K3 slots is multiplied by zero -> harmless,
//   * D rows M=2..15 are exactly zero -> lanes 16-31 of D VGPR0/1 are zero,
//     so the two 16-wide tiles merge with a plain add (no select, no EXEC
//     divergence around ds_swizzle).
__global__ __launch_bounds__(256)
void cross_stitch_wmma_kernel(const float* __restrict__ xa,
                              const float* __restrict__ xb,
                              const float* __restrict__ w,
                              float* __restrict__ oa,
                              float* __restrict__ ob)
{
    const int slab = blockIdx.x;              // n*C + c
    const int c    = slab & (kC - 1);
    const int lane = threadIdx.x & 31;
    const int wv   = threadIdx.x >> 5;        // 0..7

    // Per-channel 2x2 weights; uniform across the block -> scalar loads.
    const float w00 = w[c * 4 + 0];
    const float w01 = w[c * 4 + 1];
    const float w10 = w[c * 4 + 2];
    const float w11 = w[c * 4 + 3];

    v2f A;
    A.x = (lane == 0) ? w00 : ((lane == 1) ? w10 : 0.0f);   // K0 column
    A.y = (lane == 0) ? w01 : ((lane == 1) ? w11 : 0.0f);   // K1 column

    const bool lo = (lane < 16);
    size_t idx = (size_t)slab * kHW + (size_t)wv * 512 + lane;

    #pragma unroll 4
    for (int it = 0; it < 16; ++it, idx += 32) {
        // Full-wave 128B coalesced streaming loads.
        float va = xa[idx];
        float vb = xb[idx];

        // Prefetch ~2 iterations ahead into GL2 (global_prefetch_b8;
        // speculative, silently dropped past the allocation on the last slab).
        __builtin_prefetch(xa + idx + 64, 0, 1);
        __builtin_prefetch(xb + idx + 64, 0, 1);

        // Tile 1 (positions base+16..base+31) needs lane L <- lane L+16.
        float vah = swap16(va);
        float vbh = swap16(vb);

        // Mask B's low half explicitly (defensive vs non-finite inputs);
        // upper lanes land in K2/K3 slots which A zeroes out.
        v2f B0, B1;
        B0.x = lo ? va  : 0.0f;   // K0 row = x_a (tile 0)
        B0.y = lo ? vb  : 0.0f;   // K1 row = x_b
        B1.x = lo ? vah : 0.0f;   // tile 1
        B1.y = lo ? vbh : 0.0f;

        v8f z = {0.f, 0.f, 0.f, 0.f, 0.f, 0.f, 0.f, 0.f};
        v8f d0 = __builtin_amdgcn_wmma_f32_16x16x4_f32(
            false, A, false, B0, (short)0, z, false, false);
        v8f d1 = __builtin_amdgcn_wmma_f32_16x16x4_f32(
            false, A, false, B1, (short)0, z, false, false);

        // Branch-free half-merge: d0[*] is zero in lanes 16-31 (row M=8),
        // swap16(d1[*]) is zero in lanes 0-15. Sum = full 32-lane result.
        float ra = d0[0] + swap16(d1[0]);   // out_a
        float rb = d0[1] + swap16(d1[1]);   // out_b

        oa[idx] = ra;   // full-wave 128B coalesced stores
        ob[idx] = rb;
    }
}

extern "C" void kernel_launch(void* const* d_in, const int* in_sizes, int n_in,
                              void* d_out, int out_size, void* d_ws, size_t ws_size,
                              hipStream_t stream) {
    const float* xa = (const float*)d_in[0];
    const float* xb = (const float*)d_in[1];
    const float* w  = (const float*)d_in[2];   // [C,2,2] row-major: w00 w01 w10 w11
    float* oa = (float*)d_out;                 // outputs concatenated: out_a then out_b
    float* ob = oa + kElems;

    dim3 grid(kN * kC);    // 8192 slabs
    dim3 block(256);       // 8 wave32
    hipLaunchKernelGGL(cross_stitch_wmma_kernel, grid, block, 0, stream,
                       xa, xb, w, oa, ob);
}